// MPPContextEmbedding_67920612818961
// MI455X (gfx1250) — compile-verified
//
#include <hip/hip_runtime.h>
#include <hip/hip_bf16.h>

// out[8192,256] = concat(obs[71], latent[:,t,:]) @ W.T + b
// M=8192, N=256, K=327 padded to 328 (zero column at c=71).
// fp32 via V_WMMA_F32_16X16X4_F32. HBM floor ~18MB => ~0.8us @ 23.3 TB/s.
// A fragments register-resident per wave (reused over 4 N-tiles); B tiles
// double-buffered in LDS via global_load_async_to_lds_b128 (ASYNCcnt);
// inner loop = ds_load_b64 + wmma only.

typedef __attribute__((ext_vector_type(2))) float v2f;
typedef __attribute__((ext_vector_type(8))) float v8f;

#define KG        82                // k-groups of 4 padded columns (KPAD = 328)
#define TILE_F    (KG * 2 * 32)     // floats per staged B tile  (5248 = 21KB)
#define TILE_V2   (TILE_F / 2)      // v2f elements per tile     (2624)
#define TILE_B16  (TILE_F / 4)      // 16-byte chunks per tile   (1312)
#define WTP_FLOATS (KG * 2 * 512)   // 83,968
#define INVCAP_OFF WTP_FLOATS       // +32 floats
#define INVTLC_OFF (WTP_FLOATS + 32)// +7 floats

// ---------------------------------------------------------------------------
// Kernel 1: repack W[256,327] into fragment-native layout + reciprocal tables.
//   WtP[g][n][2], g = kgroup*2 + khalf; element j = padded column
//   c = kgroup*4 + khalf*2 + j;  c<71 -> W k=c ; c==71 -> 0 ; else W k=c-1
// ---------------------------------------------------------------------------
__global__ __launch_bounds__(256) void wt_prep_kernel(const float* __restrict__ W,
                                                      const float* __restrict__ cap,
                                                      const float* __restrict__ tlc,
                                                      float* __restrict__ ws) {
    int idx = blockIdx.x * 256 + threadIdx.x;
    if (blockIdx.x == 0) {                       // one-time reciprocal tables
        if (threadIdx.x < 32) ws[INVCAP_OFF + threadIdx.x] = 1.0f / cap[threadIdx.x];
        if (threadIdx.x < 7)  ws[INVTLC_OFF + threadIdx.x] = 1.0f / tlc[threadIdx.x];
    }
    if (idx >= WTP_FLOATS) return;
    int g     = idx >> 9;              // 0..163
    int r     = idx & 511;
    int n     = r >> 1;                // 0..255
    int j     = r & 1;
    int kg    = g >> 1;
    int khalf = g & 1;
    int c     = kg * 4 + khalf * 2 + j;
    int wk    = (c < 71) ? c : ((c == 71) ? -1 : c - 1);
    ws[idx] = (wk >= 0 && wk < 327) ? W[n * 327 + wk] : 0.0f;
}

// Per-lane observation feature for padded column c (row gr); multiply-only.
__device__ __forceinline__ float obs_feat(int c, int gr,
        const float* __restrict__ rc,   const float* __restrict__ rlc,
        const float* __restrict__ itlc, const float* __restrict__ lcg,
        const float* __restrict__ vcg,  const float* __restrict__ apol,
        const float* __restrict__ apod, const float* __restrict__ icap) {
    if (c < 32) return rc [gr * 32 +  c      ] * icap[c];
    if (c < 39) return rlc[gr * 7  + (c - 32)] * itlc[c - 32];
    if (c < 47) return lcg[gr * 8  + (c - 39)];
    if (c < 55) return vcg[gr * 8  + (c - 47)];
    if (c < 63) return apol[gr * 8 + (c - 55)] * 0.125f;   // /P, P=8
    if (c < 71) return apod[gr * 8 + (c - 63)] * 0.125f;
    return 0.0f;                                           // pad col 71
}

// ---------------------------------------------------------------------------
// Kernel 2: A-in-registers, B double-buffered in LDS via async-to-LDS DMA.
// Block = 256 thr (8 waves). Wave w owns rows [mbase + w*16, +16).
// blockIdx: 64 M-blocks x 4 N-quarters (4 N-tiles each).
// ---------------------------------------------------------------------------
__global__ __launch_bounds__(256) void ctx_gemm_kernel(
    const float* __restrict__ latent,   // [8192,50,256]
    const float* __restrict__ rc,       // [8192,32]
    const float* __restrict__ rlc,      // [8192,7]
    const float* __restrict__ lcg,      // [8192,8]
    const float* __restrict__ vcg,      // [8192,8]
    const float* __restrict__ apol,     // [8192,8]
    const float* __restrict__ apod,     // [8192,8]
    const float* __restrict__ WtP,      // fragment-native packed W (in ws)
    const float* __restrict__ icap,     // 1/capacity  [32] (in ws)
    const float* __restrict__ itlc,     // 1/target_lc [7]  (in ws)
    const float* __restrict__ bias,     // [256]
    const int*   __restrict__ timestep, // use [0]
    float*       __restrict__ out)      // [8192,256]
{
    __shared__ __align__(16) v2f Bs[2 * TILE_V2];    // 41,984 B, double buffer

    const int tid   = threadIdx.x;
    const int wave  = tid >> 5;
    const int lane  = tid & 31;
    const int lr    = lane & 15;      // A row within 16 / B,D column within 16
    const int khalf = lane >> 4;      // K-pair select per ISA 16x4 f32 layout

    const int mblk  = blockIdx.x >> 2;          // 0..63
    const int nblk  = blockIdx.x & 3;           // 0..3 -> N-tiles nblk*4..+3
    const int mbase = mblk * 128;
    const int gr    = mbase + wave * 16 + lr;   // this lane's global row
    const int t     = timestep[0];

    // ---- Build A fragments in registers: 82 x v2f, columns c = 4kg+2*khalf+{0,1}
    v2f A[KG];
    #pragma unroll
    for (int kg = 0; kg < 8; ++kg) {            // rc region: vectorized, 8B aligned
        int c  = kg * 4 + khalf * 2;
        v2f x  = *(const v2f*)(rc + gr * 32 + c);
        v2f ic = *(const v2f*)(icap + c);
        A[kg].x = x.x * ic.x;
        A[kg].y = x.y * ic.y;
    }
    #pragma unroll
    for (int kg = 8; kg < 18; ++kg) {           // boundary regions: scalar
        int c = kg * 4 + khalf * 2;
        A[kg].x = obs_feat(c,     gr, rc, rlc, itlc, lcg, vcg, apol, apod, icap);
        A[kg].y = obs_feat(c + 1, gr, rc, rlc, itlc, lcg, vcg, apol, apod, icap);
    }
    const float* lrow = latent + (gr * 50 + t) * 256;   // this lane's embedding row
    #pragma unroll
    for (int kg = 18; kg < KG; ++kg) {          // embedding region: e = c-72
        int e = (kg - 18) * 4 + khalf * 2;      // 8B-aligned
        A[kg] = *(const v2f*)(lrow + e);
    }

    // LDS byte address of Bs via ptrtoint: captures Bs (so the asm "memory"
    // clobber keeps the ds reads alive) and gives the wave-relative offset.
    const unsigned lds_base = (unsigned)(uintptr_t)(&Bs[0]);

    // ---- Stage first B tile with async-to-LDS DMA (ASYNCcnt-tracked) ----
    // Tile for ntile: 164 chunks of 128B at WtP[g*512 + ntile*32], g=0..163.
    {
        const int ntile = nblk * 4;
        for (int c = tid; c < TILE_B16; c += 256) {
            int g = c >> 3, sub = c & 7;
            unsigned long long ga =
                (unsigned long long)(uintptr_t)(WtP + g * 512 + ntile * 32 + sub * 4);
            unsigned lds_off = lds_base + (unsigned)(c * 16);
            asm volatile("global_load_async_to_lds_b128 %0, %1, off"
                         :: "v"(lds_off), "v"(ga) : "memory");
        }
    }
    asm volatile("s_wait_asynccnt 0x0" ::: "memory");
    __syncthreads();

    // ---- 4 N-tiles, double-buffered: DMA next tile while computing current ----
    for (int j = 0; j < 4; ++j) {
        const int cur = j & 1;

        if (j < 3) {                        // prefetch next tile into other buffer
            const int ntile = nblk * 4 + j + 1;
            const unsigned base = lds_base + (unsigned)((cur ^ 1) * (TILE_F * 4));
            for (int c = tid; c < TILE_B16; c += 256) {
                int g = c >> 3, sub = c & 7;
                unsigned long long ga =
                    (unsigned long long)(uintptr_t)(WtP + g * 512 + ntile * 32 + sub * 4);
                unsigned lds_off = base + (unsigned)(c * 16);
                asm volatile("global_load_async_to_lds_b128 %0, %1, off"
                             :: "v"(lds_off), "v"(ga) : "memory");
            }
        }

        // Pure LDS + WMMA inner loop; two independent accumulator chains.
        // Direct shared-array indexing -> ds_load_b64 with immediate offsets.
        const int bufv = cur * TILE_V2;
        v8f c0 = {}, c1 = {};
        #pragma unroll
        for (int kg = 0; kg < KG; kg += 2) {
            v2f b0 = Bs[bufv + (2 *  kg      + khalf) * 16 + lr];
            v2f b1 = Bs[bufv + (2 * (kg + 1) + khalf) * 16 + lr];
            c0 = __builtin_amdgcn_wmma_f32_16x16x4_f32(false, A[kg],     false, b0,
                                                       (short)0, c0, false, false);
            c1 = __builtin_amdgcn_wmma_f32_16x16x4_f32(false, A[kg + 1], false, b1,
                                                       (short)0, c1, false, false);
        }

        const int n  = (nblk * 4 + j) * 16 + lr;
        const float bb = bias[n];
        #pragma unroll
        for (int i = 0; i < 8; ++i) {
            // D layout: VGPR i -> M = i (lanes 0-15) / i+8 (lanes 16-31), N = lane%16
            int m = mbase + wave * 16 + i + 8 * khalf;
            out[m * 256 + n] = c0[i] + c1[i] + bb;
        }

        // Drain this wave's DMA, then block-wide sync before buffer swap.
        asm volatile("s_wait_asynccnt 0x0" ::: "memory");
        __syncthreads();
    }
}

extern "C" void kernel_launch(void* const* d_in, const int* in_sizes, int n_in,
                              void* d_out, int out_size, void* d_ws, size_t ws_size,
                              hipStream_t stream) {
    const float* latent = (const float*)d_in[0];
    const float* rc     = (const float*)d_in[1];
    const float* rlc    = (const float*)d_in[2];
    const float* tlc    = (const float*)d_in[3];
    const float* lcg    = (const float*)d_in[4];
    const float* vcg    = (const float*)d_in[5];
    const float* apol   = (const float*)d_in[6];
    const float* apod   = (const float*)d_in[7];
    const float* cap    = (const float*)d_in[8];
    const float* W      = (const float*)d_in[9];
    const float* bias   = (const float*)d_in[10];
    const int*   tstep  = (const int*)d_in[11];
    float*       out    = (float*)d_out;
    float*       ws     = (float*)d_ws;          // 336,028 B used

    // 1) repack W + reciprocal tables
    wt_prep_kernel<<<(WTP_FLOATS + 255) / 256, 256, 0, stream>>>(W, cap, tlc, ws);

    // 2) A-in-registers / B-in-LDS WMMA GEMM; 64 M-blocks x 4 N-quarters
    ctx_gemm_kernel<<<256, 256, 0, stream>>>(
        latent, rc, rlc, lcg, vcg, apol, apod,
        ws, ws + INVCAP_OFF, ws + INVTLC_OFF, bias, tstep, out);
}